// GCN_80977313399326
// MI455X (gfx1250) — compile-verified
//
#include <hip/hip_runtime.h>

#define N_NODES 100000
#define N_EDGES 1600000
#define NFEAT   512
#define NHID    128
#define NCLASS  40

typedef __attribute__((ext_vector_type(2))) float v2f;
typedef __attribute__((ext_vector_type(8))) float v8f;

// ---------------------------------------------------------------- utilities
__global__ void zero_i32(int* __restrict__ p, int n) {
  int i = blockIdx.x * blockDim.x + threadIdx.x;
  int stride = gridDim.x * blockDim.x;
  for (; i < n; i += stride) p[i] = 0;
}

__device__ __forceinline__ unsigned rotl32(unsigned x, int d) {
  return (x << d) | (x >> (32 - d));
}

// Exact JAX threefry2x32 bits for element i of a flat array of 2*half
// uint32s generated with key = [0, 42]  (jax.random.key(42)).
__device__ unsigned threefry_bits_key42(unsigned i, unsigned half) {
  unsigned x0, x1;
  bool lo_half;
  if (i < half) { x0 = i;        x1 = i + half; lo_half = true;  }
  else          { x0 = i - half; x1 = i;        lo_half = false; }
  const unsigned ks0 = 0u, ks1 = 42u;
  const unsigned ks2 = ks0 ^ ks1 ^ 0x1BD11BDAu;
  x0 += ks0; x1 += ks1;
#define TF_R(r) { x0 += x1; x1 = rotl32(x1, r); x1 ^= x0; }
#define TF_G0   TF_R(13) TF_R(15) TF_R(26) TF_R(6)
#define TF_G1   TF_R(17) TF_R(29) TF_R(16) TF_R(24)
  TF_G0 x0 += ks1; x1 += ks2 + 1u;
  TF_G1 x0 += ks2; x1 += ks0 + 2u;
  TF_G0 x0 += ks0; x1 += ks1 + 3u;
  TF_G1 x0 += ks1; x1 += ks2 + 4u;
  TF_G0 x0 += ks2; x1 += ks0 + 5u;
#undef TF_G1
#undef TF_G0
#undef TF_R
  return lo_half ? x0 : x1;
}

__device__ __forceinline__ float dropout_scale_key42(int i) {
  const unsigned bits = threefry_bits_key42((unsigned)i, (unsigned)(N_NODES * NHID / 2));
  const float u = __uint_as_float((bits >> 9) | 0x3F800000u) - 1.0f;
  return (u < 0.5f) ? 2.0f : 0.0f;
}

// ------------------------------------------- CSR build: histogram over dst
__global__ void edge_hist(const int* __restrict__ dst, int* __restrict__ deg) {
  int e = blockIdx.x * blockDim.x + threadIdx.x;
  int stride = gridDim.x * blockDim.x;
  for (; e < N_EDGES; e += stride) atomicAdd(&deg[dst[e]], 1);
}

// ------------------------------------------- CSR build: 2-level scan
// K1: per-block Hillis-Steele inclusive scan of 256 elems -> exclusive out
__global__ __launch_bounds__(256)
void scan_block(const int* __restrict__ in, int* __restrict__ out,
                int* __restrict__ bsum, int n) {
  __shared__ int tmp[256];
  const int tid = threadIdx.x;
  const int gi = blockIdx.x * 256 + tid;
  const int v = (gi < n) ? in[gi] : 0;
  tmp[tid] = v;
  __syncthreads();
  int acc = v;
  for (int d = 1; d < 256; d <<= 1) {
    const int t = (tid >= d) ? tmp[tid - d] : 0;
    __syncthreads();
    acc += t;
    tmp[tid] = acc;
    __syncthreads();
  }
  if (gi < n) out[gi] = acc - v;              // exclusive
  if (tid == 255) bsum[blockIdx.x] = acc;     // block total
}

// K2: scan the (<=512) block sums in one workgroup
__global__ __launch_bounds__(512)
void scan_aux(const int* __restrict__ in, int* __restrict__ out, int n) {
  __shared__ int tmp[512];
  const int tid = threadIdx.x;
  const int v = (tid < n) ? in[tid] : 0;
  tmp[tid] = v;
  __syncthreads();
  int acc = v;
  for (int d = 1; d < 512; d <<= 1) {
    const int t = (tid >= d) ? tmp[tid - d] : 0;
    __syncthreads();
    acc += t;
    tmp[tid] = acc;
    __syncthreads();
  }
  if (tid < n) out[tid] = acc - v;            // exclusive block offsets
}

// K3: add block offsets, finalize rowstart, init scatter cursor
__global__ void scan_add(int* __restrict__ rowstart, const int* __restrict__ aux,
                         int* __restrict__ cursor) {
  int i = blockIdx.x * blockDim.x + threadIdx.x;
  int stride = gridDim.x * blockDim.x;
  for (; i < N_NODES; i += stride) {
    const int r = rowstart[i] + aux[i >> 8];
    rowstart[i] = r;
    cursor[i] = r;
  }
  if (blockIdx.x == 0 && threadIdx.x == 0) rowstart[N_NODES] = N_EDGES;
}

// ------------------------------------------- CSR build: scatter edges
__global__ void csr_scatter(const int* __restrict__ src, const int* __restrict__ dst,
                            const float* __restrict__ val, int* __restrict__ cursor,
                            int* __restrict__ csrc, float* __restrict__ cval) {
  int e = blockIdx.x * blockDim.x + threadIdx.x;
  int stride = gridDim.x * blockDim.x;
  for (; e < N_EDGES; e += stride) {
    const int p = atomicAdd(&cursor[dst[e]], 1);
    csrc[p] = src[e];
    cval[p] = val[e];
  }
}

// ------------------------------------------------- GEMM1: [N,512]x[512,128]
// one wave per 16-row strip; 8 n-tiles of 16 cols; v_wmma_f32_16x16x4_f32
__global__ __launch_bounds__(256)
void gemm_x_w1(const float* __restrict__ X, const float* __restrict__ W,
               float* __restrict__ OUT) {
  const int lane  = threadIdx.x & 31;
  const int strip = blockIdx.x * 8 + (threadIdx.x >> 5);   // wave-uniform
  if (strip >= N_NODES / 16) return;                       // whole-wave exit
  const int mlo = lane & 15;
  const int hi  = lane >> 4;                               // 0 or 1
  const float* xrow = X + (long long)(strip * 16 + mlo) * NFEAT + 2 * hi;

  v8f acc[8] = {};
  for (int k = 0; k < NFEAT; k += 4) {
    v2f a;
    a.x = xrow[k];
    a.y = xrow[k + 1];
    const float* wk = W + (k + 2 * hi) * NHID + mlo;
#pragma unroll
    for (int j = 0; j < 8; ++j) {
      v2f b;
      b.x = wk[j * 16];
      b.y = wk[NHID + j * 16];
      acc[j] = __builtin_amdgcn_wmma_f32_16x16x4_f32(
          false, a, false, b, (short)0, acc[j], false, false);
    }
  }
  const int row = strip * 16 + hi * 8;
#pragma unroll
  for (int j = 0; j < 8; ++j) {
    float* o = OUT + (long long)row * NHID + j * 16 + mlo;
#pragma unroll
    for (int r = 0; r < 8; ++r) o[(long long)r * NHID] = acc[j][r];
  }
}

// ------------------------------------------------- GEMM2: [N,128]x[128,40]
__global__ __launch_bounds__(256)
void gemm_h_w2(const float* __restrict__ H, const float* __restrict__ W,
               float* __restrict__ OUT) {
  const int lane  = threadIdx.x & 31;
  const int strip = blockIdx.x * 8 + (threadIdx.x >> 5);
  if (strip >= N_NODES / 16) return;
  const int mlo = lane & 15;
  const int hi  = lane >> 4;
  const float* hrow = H + (long long)(strip * 16 + mlo) * NHID + 2 * hi;

  v8f acc[3] = {};
  for (int k = 0; k < NHID; k += 4) {
    v2f a;
    a.x = hrow[k];
    a.y = hrow[k + 1];
#pragma unroll
    for (int j = 0; j < 3; ++j) {
      const int n = j * 16 + mlo;
      v2f b;
      b.x = (n < NCLASS) ? W[(k + 2 * hi) * NCLASS + n] : 0.0f;
      b.y = (n < NCLASS) ? W[(k + 2 * hi + 1) * NCLASS + n] : 0.0f;
      acc[j] = __builtin_amdgcn_wmma_f32_16x16x4_f32(
          false, a, false, b, (short)0, acc[j], false, false);
    }
  }
  const int row = strip * 16 + hi * 8;
#pragma unroll
  for (int j = 0; j < 3; ++j) {
    const int col = j * 16 + mlo;
    if (col < NCLASS) {
#pragma unroll
      for (int r = 0; r < 8; ++r)
        OUT[(long long)(row + r) * NCLASS + col] = acc[j][r];
    }
  }
}

// --------------------------- SpMM layer 1 (CSR gather) + bias+ReLU+dropout
// one wave per destination node; lane owns 4 contiguous feats (float4)
__global__ __launch_bounds__(256)
void spmm_csr_f128_fused(const float* __restrict__ S,
                         const int* __restrict__ rowstart,
                         const int* __restrict__ csrc,
                         const float* __restrict__ cval,
                         const float* __restrict__ bias,
                         float* __restrict__ H) {
  const int lane = threadIdx.x & 31;
  const int row = blockIdx.x * 8 + (threadIdx.x >> 5);
  if (row >= N_NODES) return;
  const int p0 = rowstart[row];
  const int p1 = rowstart[row + 1];
  float4 acc = make_float4(0.f, 0.f, 0.f, 0.f);
  for (int p = p0; p < p1; ++p) {
    const int s = csrc[p];
    const float v = cval[p];
    const float4 t = *(const float4*)(S + (long long)s * NHID + lane * 4);
    acc.x += v * t.x; acc.y += v * t.y; acc.z += v * t.z; acc.w += v * t.w;
  }
  const int f = lane * 4;
  const int base = row * NHID + f;
  float4 o;
  o.x = fmaxf(acc.x + bias[f + 0], 0.f) * dropout_scale_key42(base + 0);
  o.y = fmaxf(acc.y + bias[f + 1], 0.f) * dropout_scale_key42(base + 1);
  o.z = fmaxf(acc.z + bias[f + 2], 0.f) * dropout_scale_key42(base + 2);
  o.w = fmaxf(acc.w + bias[f + 3], 0.f) * dropout_scale_key42(base + 3);
  *(float4*)(H + base) = o;
}

// ------------------- SpMM layer 2 (CSR gather) + bias + log_softmax -> out
// one wave per node; lane owns class lane, lanes 0..7 also own class 32+lane
__global__ __launch_bounds__(256)
void spmm_csr_f40_fused(const float* __restrict__ S,
                        const int* __restrict__ rowstart,
                        const int* __restrict__ csrc,
                        const float* __restrict__ cval,
                        const float* __restrict__ bias,
                        float* __restrict__ OUT) {
  const int lane = threadIdx.x & 31;
  const int row = blockIdx.x * 8 + (threadIdx.x >> 5);
  if (row >= N_NODES) return;
  const int p0 = rowstart[row];
  const int p1 = rowstart[row + 1];
  float a0 = 0.f, a1 = 0.f;
  for (int p = p0; p < p1; ++p) {
    const int s = csrc[p];
    const float v = cval[p];
    const long long so = (long long)s * NCLASS;
    a0 += v * S[so + lane];
    a1 += v * S[so + 32 + (lane & 7)];   // lanes>=8 compute discarded dups (in-range)
  }
  const bool hiv = lane < (NCLASS - 32);
  const float v0 = a0 + bias[lane];
  const float v1 = hiv ? (a1 + bias[32 + lane]) : -3.4e38f;
  float m = fmaxf(v0, v1);
#pragma unroll
  for (int o = 16; o > 0; o >>= 1) m = fmaxf(m, __shfl_xor(m, o, 32));
  float s = expf(v0 - m) + (hiv ? expf(v1 - m) : 0.f);
#pragma unroll
  for (int o = 16; o > 0; o >>= 1) s += __shfl_xor(s, o, 32);
  const float lse = m + logf(s);
  float* op = OUT + (long long)row * NCLASS;
  op[lane] = v0 - lse;
  if (hiv) op[32 + lane] = v1 - lse;
}

// ---------------------------------------------------------------- launcher
extern "C" void kernel_launch(void* const* d_in, const int* in_sizes, int n_in,
                              void* d_out, int out_size, void* d_ws, size_t ws_size,
                              hipStream_t stream) {
  (void)in_sizes; (void)n_in; (void)out_size; (void)ws_size;
  const float* x    = (const float*)d_in[0];
  const int*   esrc = (const int*)d_in[1];
  const int*   edst = (const int*)d_in[2];
  const float* ev   = (const float*)d_in[3];
  const float* w1   = (const float*)d_in[4];
  const float* b1   = (const float*)d_in[5];
  const float* w2   = (const float*)d_in[6];
  const float* b2   = (const float*)d_in[7];

  float* ws       = (float*)d_ws;
  float* support1 = ws;                                    // N*128
  float* h        = support1 + (size_t)N_NODES * NHID;     // N*128
  float* support2 = h + (size_t)N_NODES * NHID;            // N*40
  float* cval     = support2 + (size_t)N_NODES * NCLASS;   // E
  int*   csrc     = (int*)(cval + N_EDGES);                // E
  int*   rowstart = csrc + N_EDGES;                        // N+1
  int*   deg      = rowstart + (N_NODES + 1);              // N (scan input)
  int*   cursor   = deg + N_NODES;                         // N
  int*   aux      = cursor + N_NODES;                      // block sums
  int*   aux2     = aux + 512;                             // scanned offsets
  float* out      = (float*)d_out;

  const int scan_blocks = (N_NODES + 255) / 256;           // 391 (<512)
  const int edge_blocks = (N_EDGES + 255) / 256;           // 6250
  const int strips = N_NODES / 16;                         // 6250
  const int gemm_blocks = (strips + 7) / 8;
  const int node_wave_blocks = (N_NODES + 7) / 8;          // 8 waves/block

  // ---- CSR-by-destination build (deterministic per call)
  zero_i32<<<scan_blocks, 256, 0, stream>>>(deg, N_NODES);
  edge_hist<<<edge_blocks, 256, 0, stream>>>(edst, deg);
  scan_block<<<scan_blocks, 256, 0, stream>>>(deg, rowstart, aux, N_NODES);
  scan_aux<<<1, 512, 0, stream>>>(aux, aux2, scan_blocks);
  scan_add<<<scan_blocks, 256, 0, stream>>>(rowstart, aux2, cursor);
  csr_scatter<<<edge_blocks, 256, 0, stream>>>(esrc, edst, ev, cursor, csrc, cval);

  // ---- layer 1: dense WMMA GEMM, gather-SpMM fused with bias+ReLU+dropout
  gemm_x_w1<<<gemm_blocks, 256, 0, stream>>>(x, w1, support1);
  spmm_csr_f128_fused<<<node_wave_blocks, 256, 0, stream>>>(
      support1, rowstart, csrc, cval, b1, h);

  // ---- layer 2: dense WMMA GEMM, gather-SpMM fused with bias+log_softmax
  gemm_h_w2<<<gemm_blocks, 256, 0, stream>>>(h, w2, support2);
  spmm_csr_f40_fused<<<node_wave_blocks, 256, 0, stream>>>(
      support2, rowstart, csrc, cval, b2, out);
}